// MIM_72370198937938
// MI455X (gfx1250) — compile-verified
//
#include <hip/hip_runtime.h>
#include <hip/hip_bf16.h>
#include <math.h>

#define N_NODES   50000
#define N_HEDGES  10000
#define N_INC     800000
#define DIM       128
#define NEG_SLOPE 0.2f

typedef __attribute__((ext_vector_type(2))) float v2f;
typedef __attribute__((ext_vector_type(8))) float v8f;

// ---------------- init ----------------
__global__ void fill_f32(float* __restrict__ p, float v, int n) {
    int i = blockIdx.x * blockDim.x + threadIdx.x;
    if (i < n) p[i] = v;
}
__global__ void fill_u32(unsigned* __restrict__ p, unsigned v, int n) {
    int i = blockIdx.x * blockDim.x + threadIdx.x;
    if (i < n) p[i] = v;
}

// ---------------- degrees ----------------
__global__ void count_deg(const int* __restrict__ idx0, const int* __restrict__ idx1,
                          const float* __restrict__ hw,
                          float* __restrict__ cntB, float* __restrict__ degD, int E) {
    int e = blockIdx.x * blockDim.x + threadIdx.x;
    if (e >= E) return;
    int i0 = idx0[e], i1 = idx1[e];
    atomicAdd(cntB + i1, 1.0f);
    atomicAdd(degD + i0, hw[i1]);
}
__global__ void invert_safe(float* __restrict__ p, int n) {
    int i = blockIdx.x * blockDim.x + threadIdx.x;
    if (i < n) { float d = p[i]; p[i] = d > 0.0f ? 1.0f / d : 0.0f; }
}

// ---- node -> hyperedge sum (wave per edge, float4 per lane, atomic scatter) ----
__global__ void gather_scatter_sum(const int* __restrict__ idx0, const int* __restrict__ idx1,
                                   const float* __restrict__ x, float* __restrict__ he0, int E) {
    int wave = blockIdx.x * (blockDim.x >> 5) + (threadIdx.x >> 5);
    if (wave >= E) return;
    int lane = threadIdx.x & 31;
    int i0 = idx0[wave], i1 = idx1[wave];
    float4 v = *(const float4*)(x + (size_t)i0 * DIM + lane * 4);
    float* d = he0 + (size_t)i1 * DIM + lane * 4;
    atomicAdd(d + 0, v.x); atomicAdd(d + 1, v.y);
    atomicAdd(d + 2, v.z); atomicAdd(d + 3, v.w);
}
__global__ void scale_rows(float* __restrict__ P, const float* __restrict__ s, int total) {
    int i = blockIdx.x * blockDim.x + threadIdx.x;
    if (i < total) P[i] *= s[i / DIM];
}

// ---------------- fp32 WMMA GEMM: C[rows x 128] = A[rows x 128] @ W[128 x 128] ----
// rows must be a multiple of 16 (50000 and 10000 both are).
// Block: 256 threads = 8 waves. Block owns 16 rows (staged in LDS);
// wave w computes the 16x16 tile at columns [16w, 16w+16).
__global__ void gemm128_wmma(const float* __restrict__ A, const float* __restrict__ W,
                             float* __restrict__ C, int rows) {
    __shared__ float As[16 * DIM];
    const int rowBase = blockIdx.x * 16;
    const int t = threadIdx.x;
    {   // coalesced stage of 16x128 tile: 512 float4, 2 per thread
        const float4* src = (const float4*)(A + (size_t)rowBase * DIM);
        float4* dst = (float4*)As;
        dst[t]       = src[t];
        dst[t + 256] = src[t + 256];
    }
    __syncthreads();

    const int wv   = t >> 5;          // column tile
    const int lane = t & 31;
    const int lrow = lane & 15;       // A row / D col within tile
    const int koff = (lane >> 4) * 2; // which k-pair this half-wave holds
    const int col  = wv * 16 + lrow;  // global column

    v8f acc = {};
    #pragma unroll
    for (int k0 = 0; k0 < DIM; k0 += 4) {
        v2f a, b;
        // A 16x4 frag: lanes 0-15 rows, VGPR0/1 = K=koff/koff+1
        a.x = As[lrow * DIM + k0 + koff];
        a.y = As[lrow * DIM + k0 + koff + 1];
        // B 4x16 frag: lanes hold column `col`, k-pair by lane[4]
        b.x = W[(size_t)(k0 + koff)     * DIM + col];
        b.y = W[(size_t)(k0 + koff + 1) * DIM + col];
        acc = __builtin_amdgcn_wmma_f32_16x16x4_f32(
            false, a, false, b, (short)0, acc, false, false);
    }
    // D layout: VGPR r -> row = rowBase + 8*lane[4] + r, col = wv*16 + (lane&15)
    const int orow = rowBase + (lane >> 4) * 8;
    #pragma unroll
    for (int r = 0; r < 8; ++r)
        C[(size_t)(orow + r) * DIM + col] = acc[r];
}

// ---------------- per-row dot with a 128-vector (wave per row) ----------------
__global__ void rowdot(const float* __restrict__ X, const float* __restrict__ v,
                       float* __restrict__ out, int rows) {
    int wave = blockIdx.x * (blockDim.x >> 5) + (threadIdx.x >> 5);
    if (wave >= rows) return;
    int lane = threadIdx.x & 31;
    float4 a = *(const float4*)(X + (size_t)wave * DIM + lane * 4);
    float4 b = *(const float4*)(v + lane * 4);
    float s = a.x * b.x + a.y * b.y + a.z * b.z + a.w * b.w;
    #pragma unroll
    for (int m = 16; m > 0; m >>= 1) s += __shfl_xor(s, m, 32);
    if (lane == 0) out[wave] = s;
}

// ---------------- segment softmax helpers ----------------
__device__ __forceinline__ float leaky(float l) { return l > 0.0f ? l : NEG_SLOPE * l; }

__device__ __forceinline__ void atomicMaxFloatMono(unsigned* addr, float val) {
    unsigned u = __float_as_uint(val);
    u = (u & 0x80000000u) ? ~u : (u | 0x80000000u);   // monotone mapping
    atomicMax(addr, u);
}
__global__ void decode_max(const unsigned* __restrict__ u, float* __restrict__ f, int n) {
    int i = blockIdx.x * blockDim.x + threadIdx.x;
    if (i >= n) return;
    unsigned s = u[i];
    unsigned bits = (s & 0x80000000u) ? (s & 0x7FFFFFFFu) : ~s;
    f[i] = __uint_as_float(bits);
}
__global__ void edge_max(const int* __restrict__ idx0, const int* __restrict__ idx1,
                         const float* __restrict__ ai, const float* __restrict__ aj,
                         unsigned* __restrict__ nmaxU, int E) {
    int e = blockIdx.x * blockDim.x + threadIdx.x;
    if (e >= E) return;
    int i0 = idx0[e];
    atomicMaxFloatMono(nmaxU + i0, leaky(ai[i0] + aj[idx1[e]]));
}
__global__ void edge_expsum(const int* __restrict__ idx0, const int* __restrict__ idx1,
                            const float* __restrict__ ai, const float* __restrict__ aj,
                            const float* __restrict__ nmaxF, float* __restrict__ nsum, int E) {
    int e = blockIdx.x * blockDim.x + threadIdx.x;
    if (e >= E) return;
    int i0 = idx0[e];
    float l = leaky(ai[i0] + aj[idx1[e]]);
    atomicAdd(nsum + i0, __expf(l - nmaxF[i0]));
}

// ---- weighted message scatter: dst[dIdx] += scale[dIdx]*alpha(e)*src[sIdx] ----
// srcIsNode=1: src indexed by idx0, dst/scale by idx1 (node -> hyperedge)
// srcIsNode=0: src indexed by idx1, dst/scale by idx0 (hyperedge -> node)
__global__ void edge_msg(const int* __restrict__ idx0, const int* __restrict__ idx1,
                         const float* __restrict__ ai, const float* __restrict__ aj,
                         const float* __restrict__ nmaxF, const float* __restrict__ nsum,
                         const float* __restrict__ scaleArr,
                         const float* __restrict__ src, float* __restrict__ dst,
                         int srcIsNode, int E) {
    int wave = blockIdx.x * (blockDim.x >> 5) + (threadIdx.x >> 5);
    if (wave >= E) return;
    int lane = threadIdx.x & 31;
    int i0 = idx0[wave], i1 = idx1[wave];
    float l = leaky(ai[i0] + aj[i1]);
    float alpha = __expf(l - nmaxF[i0]) / nsum[i0];
    int sIdx = srcIsNode ? i0 : i1;
    int dIdx = srcIsNode ? i1 : i0;
    float s = scaleArr[dIdx] * alpha;
    float4 v = *(const float4*)(src + (size_t)sIdx * DIM + lane * 4);
    float* d = dst + (size_t)dIdx * DIM + lane * 4;
    atomicAdd(d + 0, s * v.x); atomicAdd(d + 1, s * v.y);
    atomicAdd(d + 2, s * v.z); atomicAdd(d + 3, s * v.w);
}

// ---------------- launcher ----------------
extern "C" void kernel_launch(void* const* d_in, const int* in_sizes, int n_in,
                              void* d_out, int out_size, void* d_ws, size_t ws_size,
                              hipStream_t stream) {
    const float* x    = (const float*)d_in[0];
    const int*   hidx = (const int*)d_in[1];
    const int*   idx0 = hidx;            // node ids      [E]
    const int*   idx1 = hidx + N_INC;    // hyperedge ids [E]
    const float* hw   = (const float*)d_in[2];
    const float* Wl   = (const float*)d_in[3];
    const float* Wl2  = (const float*)d_in[4];
    const float* Wl3  = (const float*)d_in[5];
    const float* att  = (const float*)d_in[6];   // [256]: [:128] node, [128:] hedge
    const float* att2 = (const float*)d_in[7];
    float* out = (float*)d_out;

    // workspace carve-up (floats)
    float* p = (float*)d_ws;
    float* xl    = p; p += (size_t)N_NODES  * DIM;
    float* hl    = p; p += (size_t)N_HEDGES * DIM;
    float* he0   = p; p += (size_t)N_HEDGES * DIM;
    float* out_e = p; p += (size_t)N_HEDGES * DIM;
    float* he2   = p; p += (size_t)N_HEDGES * DIM;
    float* Binv  = p; p += N_HEDGES;
    float* Dinv  = p; p += N_NODES;
    float* a_i   = p; p += N_NODES;
    float* a_j   = p; p += N_HEDGES;
    float* a_i2  = p; p += N_NODES;
    float* a_j2  = p; p += N_HEDGES;
    unsigned* nmaxU = (unsigned*)p; p += N_NODES;
    float* nsum  = p; p += N_NODES;
    float* nmaxF = p; p += N_NODES;

    const int T = 256;
    auto grid1 = [](int n) { return (n + 255) / 256; };
    const int WPB = T / 32;                       // waves per block
    const int gE  = (N_INC + WPB - 1) / WPB;      // wave-per-edge grid

    // ---- init (ws/out are poisoned by harness) ----
    fill_f32<<<grid1(N_HEDGES), T, 0, stream>>>(Binv, 0.0f, N_HEDGES);
    fill_f32<<<grid1(N_NODES),  T, 0, stream>>>(Dinv, 0.0f, N_NODES);
    fill_f32<<<grid1(N_HEDGES * DIM), T, 0, stream>>>(he0,   0.0f, N_HEDGES * DIM);
    fill_f32<<<grid1(N_HEDGES * DIM), T, 0, stream>>>(out_e, 0.0f, N_HEDGES * DIM);
    fill_f32<<<grid1(N_NODES * DIM),  T, 0, stream>>>(out,   0.0f, N_NODES * DIM);
    fill_u32<<<grid1(N_NODES), T, 0, stream>>>(nmaxU, 0u,   N_NODES);
    fill_f32<<<grid1(N_NODES), T, 0, stream>>>(nsum,  0.0f, N_NODES);

    // ---- degrees: Binv = 1/|e|, Dinv = 1/deg(n) ----
    count_deg<<<grid1(N_INC), T, 0, stream>>>(idx0, idx1, hw, Binv, Dinv, N_INC);
    invert_safe<<<grid1(N_HEDGES), T, 0, stream>>>(Binv, N_HEDGES);
    invert_safe<<<grid1(N_NODES),  T, 0, stream>>>(Dinv, N_NODES);

    // ---- he0 = mean of node features per hyperedge ----
    gather_scatter_sum<<<gE, T, 0, stream>>>(idx0, idx1, x, he0, N_INC);
    scale_rows<<<grid1(N_HEDGES * DIM), T, 0, stream>>>(he0, Binv, N_HEDGES * DIM);

    // ---- projections (fp32 WMMA) ----
    gemm128_wmma<<<N_NODES  / 16, T, 0, stream>>>(x,   Wl,  xl, N_NODES);
    gemm128_wmma<<<N_HEDGES / 16, T, 0, stream>>>(he0, Wl2, hl, N_HEDGES);

    // ---- attention 1: alpha = softmax_n( leaky(xl[n].att_a + hl[m].att_b) ) ----
    rowdot<<<(N_NODES  + WPB - 1) / WPB, T, 0, stream>>>(xl, att,        a_i, N_NODES);
    rowdot<<<(N_HEDGES + WPB - 1) / WPB, T, 0, stream>>>(hl, att + DIM,  a_j, N_HEDGES);
    edge_max<<<grid1(N_INC), T, 0, stream>>>(idx0, idx1, a_i, a_j, nmaxU, N_INC);
    decode_max<<<grid1(N_NODES), T, 0, stream>>>(nmaxU, nmaxF, N_NODES);
    edge_expsum<<<grid1(N_INC), T, 0, stream>>>(idx0, idx1, a_i, a_j, nmaxF, nsum, N_INC);

    // ---- msg1: out_e[m] += Binv[m]*alpha*xl[n] ----
    edge_msg<<<gE, T, 0, stream>>>(idx0, idx1, a_i, a_j, nmaxF, nsum,
                                   Binv, xl, out_e, /*srcIsNode=*/1, N_INC);

    // ---- he2 = out_e @ W_lin3, attention 2 ----
    gemm128_wmma<<<N_HEDGES / 16, T, 0, stream>>>(out_e, Wl3, he2, N_HEDGES);
    rowdot<<<(N_NODES  + WPB - 1) / WPB, T, 0, stream>>>(xl,  att2,       a_i2, N_NODES);
    rowdot<<<(N_HEDGES + WPB - 1) / WPB, T, 0, stream>>>(he2, att2 + DIM, a_j2, N_HEDGES);

    fill_u32<<<grid1(N_NODES), T, 0, stream>>>(nmaxU, 0u,   N_NODES);
    fill_f32<<<grid1(N_NODES), T, 0, stream>>>(nsum,  0.0f, N_NODES);
    edge_max<<<grid1(N_INC), T, 0, stream>>>(idx0, idx1, a_i2, a_j2, nmaxU, N_INC);
    decode_max<<<grid1(N_NODES), T, 0, stream>>>(nmaxU, nmaxF, N_NODES);
    edge_expsum<<<grid1(N_INC), T, 0, stream>>>(idx0, idx1, a_i2, a_j2, nmaxF, nsum, N_INC);

    // ---- msg2: out[n] += Dinv[n]*alpha2*out_e[m] ----
    edge_msg<<<gE, T, 0, stream>>>(idx0, idx1, a_i2, a_j2, nmaxF, nsum,
                                   Dinv, out_e, out, /*srcIsNode=*/0, N_INC);
}